// HMM_60361470378728
// MI455X (gfx1250) — compile-verified
//
#include <hip/hip_runtime.h>
#include <math.h>

#define S 32
#define D 32
#define LOG_TAU 1.8378770664093453f  // log(2*pi)

typedef __attribute__((ext_vector_type(2))) float v2f;
typedef __attribute__((ext_vector_type(8))) float v8f;

// float offsets inside d_ws
#define OFF_M      0          // S*D*D = 32768   M_k = L_k^{-1} (row-major, zero upper)
#define OFF_B      32768      // S*D             b_k = M_k * mu_k
#define OFF_CONST  33792      // S               -0.5*(D*log(2pi) + logdet_k)
#define OFF_LOGAT  33824      // S*S             logAT[j*S+i] = log_A[i][j] (transposed)
#define OFF_LOGPI  34848      // S
#define OFF_COV    34880      // S*D*D scratch
#define OFF_EM     67648      // T*S emissions

// ---------------------------------------------------------------------------
// Phase A: per-state cov -> cholesky -> triangular inverse; log-softmax of A, pi
// ---------------------------------------------------------------------------
__global__ __launch_bounds__(64) void hmm_precompute(
    const float* __restrict__ mean, const float* __restrict__ chol,
    const float* __restrict__ trans, const float* __restrict__ pi0,
    float* __restrict__ ws)
{
  const int tid = threadIdx.x;
  if (tid < S) {
    const int k = tid;
    const float* ch = chol + k * D * D;
    float* C = ws + OFF_COV + k * D * D;
    // cov = chol * chol^T + 1e-6 I
    for (int i = 0; i < D; i++)
      for (int j = 0; j < D; j++) {
        float s = 0.f;
        for (int l = 0; l < D; l++) s += ch[i * D + l] * ch[j * D + l];
        C[i * D + j] = s + (i == j ? 1e-6f : 0.f);
      }
    // in-place lower Cholesky
    float logdet = 0.f;
    for (int c = 0; c < D; c++) {
      float d = C[c * D + c];
      for (int l = 0; l < c; l++) d -= C[c * D + l] * C[c * D + l];
      d = sqrtf(d);
      C[c * D + c] = d;
      logdet += 2.f * __logf(d);
      const float inv = 1.f / d;
      for (int r = c + 1; r < D; r++) {
        float s = C[r * D + c];
        for (int l = 0; l < c; l++) s -= C[r * D + l] * C[c * D + l];
        C[r * D + c] = s * inv;
      }
    }
    // M = L^{-1} (lower), zero upper
    float* M = ws + OFF_M + k * D * D;
    for (int i = 0; i < D; i++) {
      for (int r = 0; r < i; r++) M[r * D + i] = 0.f;
      M[i * D + i] = 1.f / C[i * D + i];
      for (int r = i + 1; r < D; r++) {
        float s = 0.f;
        for (int l = i; l < r; l++) s += C[r * D + l] * M[l * D + i];
        M[r * D + i] = -s / C[r * D + r];
      }
    }
    // b_k = M * mu_k
    const float* mu = mean + k * D;
    for (int j = 0; j < D; j++) {
      float s = 0.f;
      for (int i = 0; i <= j; i++) s += M[j * D + i] * mu[i];
      ws[OFF_B + k * D + j] = s;
    }
    ws[OFF_CONST + k] = -0.5f * ((float)D * LOG_TAU + logdet);
  } else if (tid < 2 * S) {
    const int r = tid - S;
    const float* row = trans + r * S;
    float m = row[0];
    for (int c = 1; c < S; c++) m = fmaxf(m, row[c]);
    float s = 0.f;
    for (int c = 0; c < S; c++) s += __expf(row[c] - m);
    const float lse = m + __logf(s);
    for (int c = 0; c < S; c++) ws[OFF_LOGAT + c * S + r] = row[c] - lse;  // transposed
    if (r == 0) {
      float m2 = pi0[0];
      for (int c = 1; c < S; c++) m2 = fmaxf(m2, pi0[c]);
      float s2 = 0.f;
      for (int c = 0; c < S; c++) s2 += __expf(pi0[c] - m2);
      const float lse2 = m2 + __logf(s2);
      for (int c = 0; c < S; c++) ws[OFF_LOGPI + c] = pi0[c] - lse2;
    }
  }
}

// ---------------------------------------------------------------------------
// Phase B: emissions via f32 WMMA.  Wave w handles 16 obs rows; block stages
// M_k in LDS per state; Y = O(16x32) * M_k^T via 8x V_WMMA_F32_16X16X4_F32
// per 16x16 output tile, then quad = ||Y - b_k||^2 reduced with shfl_xor.
// ---------------------------------------------------------------------------
__global__ __launch_bounds__(256) void hmm_emissions(
    const float* __restrict__ obs, float* __restrict__ ws, int T)
{
  __shared__ float sM[D * D];
  __shared__ float sB[S];
  __shared__ float sConst;

  const int lane = threadIdx.x & 31;
  const int wave = threadIdx.x >> 5;
  const int tile = blockIdx.x * 8 + wave;    // 16 rows per tile
  const int t0   = tile * 16;
  const int r    = lane & 15;                // row in A-tile / col in B-tile
  const int h    = lane >> 4;                // lane-half -> K-pair {0,1} vs {2,3}
  const bool active = (t0 + 16) <= T;

  // A operand: 16x4 f32 layout => lane holds A[r][4kk+2h], A[r][4kk+2h+1]
  const float* orow = obs + (size_t)(active ? (t0 + r) : 0) * D + 2 * h;
  v2f a[8];
#pragma unroll
  for (int kk = 0; kk < 8; kk++) a[kk] = *(const v2f*)(orow + 4 * kk);

  const float* Mbase = ws + OFF_M;
  const float* bbase = ws + OFF_B;
  const float* cbase = ws + OFF_CONST;
  float* em = ws + OFF_EM;

  for (int k = 0; k < S; k++) {
    __syncthreads();
    {  // stage M_k (1024 floats) cooperatively
      const float4* src = (const float4*)(Mbase + k * D * D);
      ((float4*)sM)[threadIdx.x] = src[threadIdx.x];
      if (threadIdx.x < S) sB[threadIdx.x] = bbase[k * D + threadIdx.x];
      if (threadIdx.x == 0) sConst = cbase[k];
    }
    __syncthreads();

    v8f acc0 = {0.f, 0.f, 0.f, 0.f, 0.f, 0.f, 0.f, 0.f};
    v8f acc1 = {0.f, 0.f, 0.f, 0.f, 0.f, 0.f, 0.f, 0.f};
#pragma unroll
    for (int kk = 0; kk < 8; kk++) {
      const int col = 4 * kk + 2 * h;  // B[kb][n] = M_k[n][K], lane-half K split
      v2f b0 = *(const v2f*)&sM[r * D + col];          // out cols 0..15
      v2f b1 = *(const v2f*)&sM[(r + 16) * D + col];   // out cols 16..31
      acc0 = __builtin_amdgcn_wmma_f32_16x16x4_f32(false, a[kk], false, b0,
                                                   (short)0, acc0, false, false);
      acc1 = __builtin_amdgcn_wmma_f32_16x16x4_f32(false, a[kk], false, b1,
                                                   (short)0, acc1, false, false);
    }

    const float bk0 = sB[r];
    const float bk1 = sB[r + 16];
    const float cst = sConst;
    float p[8];
#pragma unroll
    for (int q = 0; q < 8; q++) {
      const float z0 = acc0[q] - bk0;
      const float z1 = acc1[q] - bk1;
      p[q] = z0 * z0 + z1 * z1;
    }
    // reduce across the 16 lanes of each half (cols live across lanes)
#pragma unroll
    for (int mask = 1; mask < 16; mask <<= 1) {
#pragma unroll
      for (int q = 0; q < 8; q++) p[q] += __shfl_xor(p[q], mask, 32);
    }
    if (active && r == 0) {
      const int rowbase = t0 + h * 8;  // lanes 0-15 hold rows 0..7, 16-31 rows 8..15
#pragma unroll
      for (int q = 0; q < 8; q++)
        em[(size_t)(rowbase + q) * S + k] = cst - 0.5f * p[q];
    }
  }
}

// ---------------------------------------------------------------------------
// Phase C: sequential soft-Viterbi scan (non-associative -> serial).
// One wave32; lane j owns state-column j.  Previous delta vector broadcast
// through LDS (1 ds_store + 8 ds_load_b128), output softmax via shfl_xor
// butterflies, emissions software-pipelined + prefetched.
// ---------------------------------------------------------------------------
__global__ __launch_bounds__(32) void hmm_scan(
    const float* __restrict__ ws, float* __restrict__ out, int T)
{
  __shared__ float sdn[S];
  const int j = threadIdx.x;
  const float* em    = ws + OFF_EM;
  const float* logAT = ws + OFF_LOGAT;

  float Acol[S];  // log_A[:, j]
#pragma unroll
  for (int i = 0; i < S; i++) Acol[i] = logAT[j * S + i];

  float dn = ws[OFF_LOGPI + j] + em[j];

  {  // output row 0: softmax across lanes
    float mm = dn;
#pragma unroll
    for (int mask = 1; mask < 32; mask <<= 1) mm = fmaxf(mm, __shfl_xor(mm, mask, 32));
    float ee = __expf(dn - mm);
    float ss = ee;
#pragma unroll
    for (int mask = 1; mask < 32; mask <<= 1) ss += __shfl_xor(ss, mask, 32);
    out[j] = ee / ss;
  }

  sdn[j] = dn;
  __syncthreads();

  float em_next = em[S + j];

  for (int t = 1; t < T; t++) {
    const float emj = em_next;
    if (t + 1 < T) em_next = em[(size_t)(t + 1) * S + j];
    if (t + 32 < T) __builtin_prefetch(&em[(size_t)(t + 32) * S + j], 0, 1);

    // broadcast previous deltas from LDS
    float dp[S];
    const float4* s4 = (const float4*)sdn;
#pragma unroll
    for (int q = 0; q < 8; q++) {
      const float4 d4 = s4[q];
      dp[4 * q + 0] = d4.x; dp[4 * q + 1] = d4.y;
      dp[4 * q + 2] = d4.z; dp[4 * q + 3] = d4.w;
    }
    __syncthreads();  // everyone read before overwrite

    float v[S];
    float m = dp[0] + Acol[0];
    v[0] = m;
#pragma unroll
    for (int i = 1; i < S; i++) {
      const float vv = dp[i] + Acol[i];
      v[i] = vv;
      m = fmaxf(m, vv);
    }
    float s = 0.f, sv = 0.f;
#pragma unroll
    for (int i = 0; i < S; i++) {
      const float e = __expf(v[i] - m);
      s += e;
      sv = fmaf(v[i], e, sv);
    }
    dn = sv / s + emj;  // TEMPERATURE == 1
    sdn[j] = dn;

    // output softmax across lanes
    float mm = dn;
#pragma unroll
    for (int mask = 1; mask < 32; mask <<= 1) mm = fmaxf(mm, __shfl_xor(mm, mask, 32));
    float ee = __expf(dn - mm);
    float ss = ee;
#pragma unroll
    for (int mask = 1; mask < 32; mask <<= 1) ss += __shfl_xor(ss, mask, 32);
    out[(size_t)t * S + j] = ee / ss;

    __syncthreads();
  }
}

// ---------------------------------------------------------------------------
extern "C" void kernel_launch(void* const* d_in, const int* in_sizes, int n_in,
                              void* d_out, int out_size, void* d_ws, size_t ws_size,
                              hipStream_t stream)
{
  const float* obs   = (const float*)d_in[0];
  const float* mean  = (const float*)d_in[1];
  const float* chol  = (const float*)d_in[2];
  const float* trans = (const float*)d_in[3];
  const float* pi0   = (const float*)d_in[4];
  float* ws  = (float*)d_ws;
  float* out = (float*)d_out;
  const int T = in_sizes[0] / D;

  hipLaunchKernelGGL(hmm_precompute, dim3(1), dim3(64), 0, stream,
                     mean, chol, trans, pi0, ws);

  const int tiles  = T / 16;            // T = 131072 -> 8192 tiles
  const int blocks = (tiles + 7) / 8;   // 8 waves (tiles) per block
  hipLaunchKernelGGL(hmm_emissions, dim3(blocks), dim3(256), 0, stream,
                     obs, ws, T);

  hipLaunchKernelGGL(hmm_scan, dim3(1), dim3(32), 0, stream, ws, out, T);
}